// SimpleGeometricEnergy_32109175505694
// MI455X (gfx1250) — compile-verified
//
#include <hip/hip_runtime.h>
#include <stdint.h>

typedef float v2f __attribute__((ext_vector_type(2)));
typedef float v8f __attribute__((ext_vector_type(8)));

#define BATCH   64
#define NNODES  65536
#define SEG     16                    // segments per batch
#define CHUNK   (NNODES / SEG)        // 4096 rows per workgroup
#define TILE    256                   // rows per LDS buffer
#define NT      (CHUNK / TILE)        // 16 tiles per workgroup
#define THREADS 256                   // 8 wave32s

// Issue one async 16B global->LDS copy (GV mode, 64-bit vaddr).
__device__ __forceinline__ void async_ld16(unsigned int lds_off, unsigned long long gaddr) {
  asm volatile("global_load_async_to_lds_b128 %0, %1, off"
               :: "v"(lds_off), "v"(gaddr) : "memory");
}

__global__ __launch_bounds__(THREADS)
void gram_wmma_kernel(const float* __restrict__ h,
                      const float* __restrict__ Wm,
                      float* __restrict__ ws) {
  // Each row stored as 16 floats: [0..7] = data, [8..15] = permanent zero pad.
  __shared__ float buf[2][TILE * 16];
  __shared__ float sred[8];

  const int tid  = threadIdx.x;
  const int lane = tid & 31;
  const int wv   = tid >> 5;
  const int wg   = blockIdx.x;          // 0..1023
  const int b    = wg >> 4;
  const int s    = wg & (SEG - 1);
  const float* seg = h + ((size_t)b * NNODES + (size_t)s * CHUNK) * 8;

  // Zero the pad half (cols 8..15) of every row in both buffers (once).
  for (int r = tid; r < 2 * TILE; r += THREADS) {
    float* row = &buf[(r >= TILE) ? 1 : 0][(r & (TILE - 1)) * 16 + 8];
#pragma unroll
    for (int i = 0; i < 8; ++i) row[i] = 0.0f;
  }
  __syncthreads();

  // Prologue: async-load tile 0 into buf[0]. 2 x b128 per thread per tile.
#pragma unroll
  for (int qq = 0; qq < 2; ++qq) {
    const int e = tid + qq * THREADS;          // 0..511 16B-chunks
    const int row = e >> 1, half = e & 1;
    async_ld16((unsigned int)(uintptr_t)&buf[0][row * 16 + half * 4],
               (unsigned long long)(uintptr_t)(seg + (size_t)row * 8 + half * 4));
  }

  v8f acc = {0.f, 0.f, 0.f, 0.f, 0.f, 0.f, 0.f, 0.f};
  const int j  = lane & 15;                    // column (0..15; >=8 reads zero pad)
  const int cc = lane >> 4;                    // K-half selector

  for (int t = 0; t < NT; ++t) {
    const int p = t & 1;
    if (t + 1 < NT) {
      // Prefetch next tile into the other buffer (it was fully consumed and
      // released by the trailing barrier of the previous iteration).
#pragma unroll
      for (int qq = 0; qq < 2; ++qq) {
        const int e = tid + qq * THREADS;
        const int row = e >> 1, half = e & 1;
        async_ld16((unsigned int)(uintptr_t)&buf[p ^ 1][row * 16 + half * 4],
                   (unsigned long long)(uintptr_t)
                       (seg + (size_t)(t + 1) * TILE * 8 + (size_t)row * 8 + half * 4));
      }
      asm volatile("s_wait_asynccnt 0x2" ::: "memory");  // tile t complete, t+1 in flight
    } else {
      asm volatile("s_wait_asynccnt 0x0" ::: "memory");  // drain last tile
    }
    __syncthreads();

    // Gram accumulate: D += A*B with A==B register contents.
    // Lane (j, cc) supplies H[k0+2cc][j] and H[k0+2cc+1][j].
    const float* base = &buf[p][(wv * 32) * 16];   // this wave's 32 rows
#pragma unroll
    for (int k0 = 0; k0 < 32; k0 += 4) {
      const int idx = (k0 + 2 * cc) * 16 + j;
      v2f a;
      a.x = base[idx];
      a.y = base[idx + 16];
      acc = __builtin_amdgcn_wmma_f32_16x16x4_f32(false, a, false, a,
                                                  (short)0, acc, false, false);
    }
    __syncthreads();   // release buf[p] for the next prefetch
  }

  // acc: lanes 0..7 hold Gram column j, rows 0..7 in acc[0..7]. Contract with
  // M = 0.5*I + 0.1*W, reduce across lanes, then across the 8 waves.
  float qv = 0.0f;
  if (lane < 8) {
#pragma unroll
    for (int i = 0; i < 8; ++i) {
      const float coef = 0.1f * Wm[i * 8 + lane] + ((i == lane) ? 0.5f : 0.0f);
      qv = fmaf(coef, acc[i], qv);
    }
  }
#pragma unroll
  for (int m = 16; m >= 1; m >>= 1) qv += __shfl_xor(qv, m, 32);
  if (lane == 0) sred[wv] = qv;
  __syncthreads();
  if (tid == 0) {
    float tot = 0.0f;
#pragma unroll
    for (int i = 0; i < 8; ++i) tot += sred[i];
    ws[wg] = tot;   // one partial per (batch, segment)
  }
}

// Deterministic fixed-order final reduction: out[b] = sum of 16 segment partials.
__global__ void seg_reduce_kernel(const float* __restrict__ ws, float* __restrict__ out) {
  const int b = threadIdx.x;
  if (b < BATCH) {
    float tot = 0.0f;
#pragma unroll
    for (int s = 0; s < SEG; ++s) tot += ws[b * SEG + s];
    out[b] = tot;
  }
}

extern "C" void kernel_launch(void* const* d_in, const int* in_sizes, int n_in,
                              void* d_out, int out_size, void* d_ws, size_t ws_size,
                              hipStream_t stream) {
  (void)in_sizes; (void)n_in; (void)out_size; (void)ws_size;
  const float* h  = (const float*)d_in[0];
  const float* Wm = (const float*)d_in[1];
  float* out = (float*)d_out;
  float* ws  = (float*)d_ws;   // needs BATCH*SEG = 1024 floats

  gram_wmma_kernel<<<BATCH * SEG, THREADS, 0, stream>>>(h, Wm, ws);
  seg_reduce_kernel<<<1, 64, 0, stream>>>(ws, out);
}